// NGram_63754494542180
// MI455X (gfx1250) — compile-verified
//
#include <hip/hip_runtime.h>
#include <cstdint>

// in  : (B=8, S=2048, E=512) f32
// out : (B, N=5, S, E) f32, out[b,k,s,:] = in[b, s+k-2, :] (zero-padded)
#define BATCH    8
#define SEQ      2048
#define E4       128            // float4 per row (512 floats, 2048 B)
#define NGRAM    5
#define LEFT     2
#define TROWSO   16             // output s-rows per block
#define TROWSL   (TROWSO + 4)   // LDS tile rows (2-row halo each side)
#define NTHREADS 256            // 8 wave32

__global__ __launch_bounds__(NTHREADS) void
ngram_async_kernel(const float4* __restrict__ in, float4* __restrict__ out) {
    // 20 rows * 2048 B = 40 KB LDS tile -> 8 blocks/WGP by LDS
    __shared__ float4 tile[TROWSL * E4];

    const unsigned tid = threadIdx.x;
    const unsigned blk = blockIdx.x;          // [0, 8*128)
    const unsigned b   = blk >> 7;            // / (SEQ/TROWSO)
    const unsigned s0  = (blk & 127u) * TROWSO;

    // Flat->LDS keeps the in-allocation byte offset in addr[31:0] (ISA 10.2).
    const unsigned ldsBase = (unsigned)(uintptr_t)(&tile[0]);

    const bool first = (s0 == 0);
    const bool last  = (s0 == SEQ - TROWSO);

    // ---- 1) zero out-of-range halo rows (2 rows = 256 float4 = 1/thread) ---
    if (first) {
        tile[tid] = float4{0.f, 0.f, 0.f, 0.f};                     // LDS rows 0,1
    } else if (last) {
        tile[(TROWSL - 2) * E4 + tid] = float4{0.f, 0.f, 0.f, 0.f}; // rows 18,19
    }

    // ---- 2) async global -> LDS, one base address + immediate offsets ------
    // IOFFSET is added to BOTH the LDS and global address (ISA 08 §4.4), and
    // within the copy both advance by 4096 B per 256-thread step.
    const unsigned lrStart = first ? 2u : 0u;          // first valid LDS row
    const int      gRow0   = (int)s0 - LEFT + (int)lrStart;
    const float4* __restrict__ gsrc =
        in + ((size_t)b * SEQ + (size_t)gRow0) * E4 + tid;
    const unsigned ldsL = ldsBase + (lrStart * E4 + tid) * 16u;

#define ALOAD(OFF)                                                             \
    asm volatile("global_load_async_to_lds_b128 %0, %1, off offset:" #OFF      \
                 :: "v"(ldsL), "v"(gsrc) : "memory")
    ALOAD(0);     ALOAD(4096);  ALOAD(8192);  ALOAD(12288); ALOAD(16384);
    ALOAD(20480); ALOAD(24576); ALOAD(28672); ALOAD(32768);       // 9 steps
    if (!first && !last) ALOAD(36864);   // interior tile: 10th step (20 rows)
#undef ALOAD

    asm volatile("s_wait_asynccnt 0x0" ::: "memory");  // own loads done
    __syncthreads();                                   // whole tile visible

    // ---- 3) fan out 5x: async LDS -> global, non-temporal streaming stores -
    // Output row (k,s) is LDS row (s - s0 + k) -> flat float4 index i + k*128.
#pragma unroll
    for (int k = 0; k < NGRAM; ++k) {
        float4* __restrict__ gdst =
            out + (((size_t)b * NGRAM + (size_t)k) * SEQ + s0) * E4 + tid;
        const unsigned ldsK = ldsBase + ((unsigned)k * E4 + tid) * 16u;
#define ASTORE(OFF)                                                            \
        asm volatile("global_store_async_from_lds_b128 %0, %1, off offset:" #OFF \
                     " th:TH_STORE_NT"                                         \
                     :: "v"(gdst), "v"(ldsK) : "memory")
        ASTORE(0);     ASTORE(4096);  ASTORE(8192);  ASTORE(12288);
        ASTORE(16384); ASTORE(20480); ASTORE(24576); ASTORE(28672); // 16 rows
#undef ASTORE
    }

    // Drain async stores before wave retire (ENDPGM also waits-idle).
    asm volatile("s_wait_asynccnt 0x0" ::: "memory");
}

extern "C" void kernel_launch(void* const* d_in, const int* in_sizes, int n_in,
                              void* d_out, int out_size, void* d_ws, size_t ws_size,
                              hipStream_t stream) {
    (void)in_sizes; (void)n_in; (void)out_size; (void)d_ws; (void)ws_size;
    const float4* in  = (const float4*)d_in[0];
    float4*       out = (float4*)d_out;

    const int blocks = BATCH * (SEQ / TROWSO);   // 1024 blocks x 256 threads
    ngram_async_kernel<<<dim3(blocks), dim3(NTHREADS), 0, stream>>>(in, out);
}